// SimpleGCNLayer_29222957482711
// MI455X (gfx1250) — compile-verified
//
#include <hip/hip_runtime.h>
#include <hip/hip_bf16.h>

#define N_NODES 16384
#define D_MODEL 128
#define N_EDGES 524288
#define WORDS_PER_ROW (N_NODES / 32)   // 512 u32 per adjacency row

typedef __attribute__((ext_vector_type(2))) float v2f;
typedef __attribute__((ext_vector_type(8))) float v8f;

// ---------------------------------------------------------------------------
// Kernel 1: build adjacency bitmask with set semantics (duplicates collapse).
// mask bit (src*N_NODES + dst) = 1
// ---------------------------------------------------------------------------
__global__ void gcn_build_mask(const long long* __restrict__ edge_index,
                               unsigned int* __restrict__ mask) {
    int e = blockIdx.x * blockDim.x + threadIdx.x;
    if (e >= N_EDGES) return;
    long long s = edge_index[e];             // edge_index[0, e]
    long long d = edge_index[N_EDGES + e];   // edge_index[1, e]
    s = s < 0 ? 0 : (s > N_NODES - 1 ? N_NODES - 1 : s);
    d = d < 0 ? 0 : (d > N_NODES - 1 ? N_NODES - 1 : d);
    unsigned int pos = (unsigned int)s * N_NODES + (unsigned int)d;  // < 2^28
    atomicOr(&mask[pos >> 5], 1u << (pos & 31));
}

// ---------------------------------------------------------------------------
// Kernel 2: h[row, :] = (1/deg(row)) * sum_{dst in row} x[dst, :]
// One block (128 threads) per src row; thread j owns feature column j.
// Mask-word loads are wave-uniform (scalar path); x loads are 128-lane
// coalesced 512B transactions that hit L2 (x is only 8MB).
// ---------------------------------------------------------------------------
__global__ void gcn_aggregate(const float* __restrict__ x,
                              const unsigned int* __restrict__ mask,
                              float* __restrict__ h) {
    const int row = blockIdx.x;       // 0..16383
    const int j   = threadIdx.x;      // 0..127 feature column
    const unsigned int* mrow = mask + (size_t)row * WORDS_PER_ROW;

    float acc = 0.0f;
    int   cnt = 0;
    for (int w = 0; w < WORDS_PER_ROW; ++w) {
        unsigned int m = mrow[w];     // uniform address -> s_load broadcast
        cnt += __popc(m);
        while (m) {
            int bit = __ffs(m) - 1;
            m &= m - 1;
            int dst = (w << 5) + bit;
            acc += x[(size_t)dst * D_MODEL + j];
        }
    }
    float deg = (float)(cnt < 1 ? 1 : cnt);
    h[(size_t)row * D_MODEL + j] = acc / deg;
}

// ---------------------------------------------------------------------------
// Kernel 3: out = h @ W^T + b  via V_WMMA_F32_16X16X4_F32.
// One wave32 per 16x16 output tile; K-loop of 32 WMMAs over D_MODEL=128.
//   A (16x4):  lane L -> M = L%16, VGPR v -> K = k + 2*(L/16) + v
//   B (4x16):  lane L -> N = L%16, VGPR v -> K = k + 2*(L/16) + v
//              B[kk, nn] = W^T[kk, j0+nn] = W[j0+nn, kk]  (row-major W read)
//   C/D:       lane L, VGPR v -> M = v + 8*(L/16), N = L%16
// ---------------------------------------------------------------------------
__global__ void gcn_gemm_wmma(const float* __restrict__ h,
                              const float* __restrict__ W,
                              const float* __restrict__ b,
                              float* __restrict__ out) {
    const int wave  = threadIdx.x >> 5;               // 4 waves per block
    const int lane  = threadIdx.x & 31;
    const int tile  = blockIdx.x * 4 + wave;          // 8192 tiles total
    const int i0    = (tile >> 3) << 4;               // row-tile * 16
    const int j0    = (tile & 7) << 4;                // col-tile * 16

    const int half = lane >> 4;                       // 0 or 1
    const int idx  = lane & 15;

    const float* hrow = h + (size_t)(i0 + idx) * D_MODEL + half * 2;
    const float* wrow = W + (size_t)(j0 + idx) * D_MODEL + half * 2;

    v8f c = {};
#pragma unroll 8
    for (int k = 0; k < D_MODEL; k += 4) {
        v2f a  = *(const v2f*)(hrow + k);
        v2f bm = *(const v2f*)(wrow + k);
        // D = A(16x4) x B(4x16) + C ; 8 args: (negA, A, negB, B, cmod, C, reuseA, reuseB)
        c = __builtin_amdgcn_wmma_f32_16x16x4_f32(
                false, a, false, bm, (short)0, c, false, false);
    }

    const float bias = b[j0 + idx];
    float* ocol = out + (size_t)(i0 + half * 8) * D_MODEL + j0 + idx;
#pragma unroll
    for (int v = 0; v < 8; ++v) {
        ocol[(size_t)v * D_MODEL] = c[v] + bias;
    }
}

// ---------------------------------------------------------------------------
extern "C" void kernel_launch(void* const* d_in, const int* in_sizes, int n_in,
                              void* d_out, int out_size, void* d_ws, size_t ws_size,
                              hipStream_t stream) {
    const float*     x  = (const float*)d_in[0];
    const long long* ei = (const long long*)d_in[1];   // int64 [2, N_EDGES]
    const float*     W  = (const float*)d_in[2];
    const float*     b  = (const float*)d_in[3];
    float*           out = (float*)d_out;

    const size_t mask_bytes = (size_t)N_NODES * WORDS_PER_ROW * sizeof(unsigned int); // 32 MB
    unsigned int* mask = (unsigned int*)d_ws;
    float*        h    = (float*)((char*)d_ws + mask_bytes);                          // 8 MB

    // Phase 0: clear adjacency bitmask (graph-capture-safe async memset)
    hipMemsetAsync(mask, 0, mask_bytes, stream);

    // Phase 1: edges -> bitmask (atomicOr gives exact "set" dedupe semantics)
    gcn_build_mask<<<(N_EDGES + 255) / 256, 256, 0, stream>>>(ei, mask);

    // Phase 2: normalized aggregation h = (adj/deg) @ x
    gcn_aggregate<<<N_NODES, D_MODEL, 0, stream>>>(x, mask, h);

    // Phase 3: out = h @ W^T + b   (WMMA f32 16x16x4)
    // 8192 tiles, 4 waves per 128-thread block -> 2048 blocks
    gcn_gemm_wmma<<<2048, 128, 0, stream>>>(h, W, b, out);
}